// IPM2_43868795961571
// MI455X (gfx1250) — compile-verified
//
#include <hip/hip_runtime.h>
#include <hip/hip_bf16.h>
#include <math.h>

typedef float v2f __attribute__((ext_vector_type(2)));
typedef float v8f __attribute__((ext_vector_type(8)));

constexpr int   N70  = 70;
constexpr int   NP   = 80;          // padded small-matrix dim
constexpr int   NPP  = NP * NP;     // 6400
constexpr int   NT   = 2485;
constexpr int   MD   = 2000;
constexpr int   CHUNK = 125;        // 2000 / 125 = 16 chunks
constexpr float SQ2F = 1.41421356237309504880f;
constexpr float SQHF = 0.70710678118654752440f;

// scalar slots in workspace
enum { S_MU = 0, S_DOTXZ, S_ALPHA, S_BETA, S_MINAB, S_E, S_G2,
       S_DOTNEW, S_SIGMA, S_SMU, S_ALPHA2, S_BETA2 };

// ---- workspace layout (in floats) ----
constexpr size_t OFF_SM  = 64;                       // 20 small 80x80 mats
constexpr size_t OFF_VEC = OFF_SM + 20ull * NPP;
constexpr size_t V_RP  = OFF_VEC;
constexpr size_t V_RD  = V_RP  + MD;
constexpr size_t V_RC  = V_RD  + NT;   // reused for Rc2
constexpr size_t V_U1  = V_RC  + NT;
constexpr size_t V_U2  = V_U1  + NT;
constexpr size_t V_U3  = V_U2  + NT;
constexpr size_t V_U4  = V_U3  + NT;
constexpr size_t V_U5  = V_U4  + NT;
constexpr size_t V_TV  = V_U5  + NT;
constexpr size_t V_CZ  = V_TV  + NT;
constexpr size_t V_H   = V_CZ  + NT;
constexpr size_t V_H2  = V_H   + MD;
constexpr size_t V_DY  = V_H2  + MD;
constexpr size_t V_DY2 = V_DY  + MD;
constexpr size_t V_YT  = V_DY2 + MD;
constexpr size_t V_DZ  = V_YT  + MD;
constexpr size_t V_DZ2 = V_DZ  + NT;
constexpr size_t V_DX  = V_DZ2 + NT;
constexpr size_t V_DX2 = V_DX  + NT;
constexpr size_t OFF_SK = V_DX2 + NT;                  // SK1 [NT x MD]
constexpr size_t OFF_MM = OFF_SK + (size_t)NT * MD;    // Mmat [MD x MD]
constexpr size_t OFF_SB = OFF_MM + (size_t)MD * MD;    // batch smat
constexpr size_t OFF_UB = OFF_SB + (size_t)CHUNK * NPP;
constexpr size_t OFF_WB = OFF_UB + (size_t)CHUNK * NPP;

// small-matrix slot indices
enum { MX = 0, MZ, MQ, MP_, MQI, MPI, MQIT, MPIT, MZI, MXZ,
       MT1, MW1, MD2, MW2, MB, MS, MT2 };

__device__ __forceinline__ void tril_ij(int t, int& i, int& j) {
    float f = sqrtf(8.0f * (float)t + 1.0f);
    int r = (int)((f - 1.0f) * 0.5f);
    while ((r + 1) * (r + 2) / 2 <= t) ++r;
    while (r * (r + 1) / 2 > t) --r;
    i = r; j = t - r * (r + 1) / 2;
}

__device__ __forceinline__ v8f wmma4(v2f a, v2f b, v8f c) {
    return __builtin_amdgcn_wmma_f32_16x16x4_f32(false, a, false, b,
                                                 (short)0, c, false, false);
}

// ---- CDNA5 async global->LDS copy (ASYNCcnt-tracked), ISA 15.18.3 op 96 ----
// LDS offset taken from the VDST VGPR; global address from a VGPR pair (SADDR off)
__device__ __forceinline__ void async_g2l_b32(unsigned ldsOff, const void* gaddr) {
    asm volatile("global_load_async_to_lds_b32 %0, %1, off"
                 :: "v"(ldsOff), "v"(gaddr) : "memory");
}
__device__ __forceinline__ void wait_async0() {
    asm volatile("s_wait_asynccnt 0x0" ::: "memory");
}
__device__ __forceinline__ unsigned lds_off(const void* p) {
    // generic pointer to LDS: low 32 bits are the LDS byte offset
    return (unsigned)(size_t)p;
}

// ---------------- small-matrix kernels ----------------

// smat: packed lower-tri vec (2485) -> symmetric 80x80 (pad diag = diagpad)
__global__ void k_smat80(const float* __restrict__ v, float* __restrict__ Mo,
                         float diagpad) {
    int idx = blockIdx.x * blockDim.x + threadIdx.x;
    if (idx >= NPP) return;
    int i = idx / NP, j = idx % NP;
    float val;
    if (i < N70 && j < N70) {
        int a = i > j ? i : j, b = i > j ? j : i;
        float x = v[a * (a + 1) / 2 + b];
        val = (i == j) ? x : SQHF * x;
    } else {
        val = (i == j) ? diagpad : 0.0f;
    }
    Mo[idx] = val;
}

__global__ void k_tr80(const float* __restrict__ in, float* __restrict__ out) {
    int idx = blockIdx.x * blockDim.x + threadIdx.x;
    if (idx >= NPP) return;
    int i = idx / NP, j = idx % NP;
    out[j * NP + i] = in[idx];
}

// upper Cholesky factor Q of SPD 80x80 (X = Q^T Q), in-LDS
__global__ __launch_bounds__(256)
void k_chol_upper80(const float* __restrict__ Ain, float* __restrict__ Qout) {
    __shared__ float T[NP][NP + 1];
    for (int e = threadIdx.x; e < NPP; e += 256) T[e / NP][e % NP] = Ain[e];
    __syncthreads();
    for (int k = 0; k < NP; ++k) {
        if (threadIdx.x == 0) T[k][k] = sqrtf(fmaxf(T[k][k], 1e-30f));
        __syncthreads();
        float dk = T[k][k];
        for (int j = k + 1 + threadIdx.x; j < NP; j += 256) T[k][j] /= dk;
        __syncthreads();
        for (int e = threadIdx.x; e < NPP; e += 256) {
            int i = e / NP, j = e % NP;
            if (i > k && j >= i) T[i][j] -= T[k][i] * T[k][j];
        }
        __syncthreads();
    }
    for (int e = threadIdx.x; e < NPP; e += 256) {
        int i = e / NP, j = e % NP;
        Qout[e] = (j >= i) ? T[i][j] : 0.0f;
    }
}

// invert upper triangular 80x80; one column per thread
__global__ __launch_bounds__(128)
void k_trinv_upper80(const float* __restrict__ Q, float* __restrict__ R) {
    __shared__ float U[NP][NP + 1];
    __shared__ float Xc[NP][NP + 1];
    for (int e = threadIdx.x; e < NPP; e += 128) U[e / NP][e % NP] = Q[e];
    __syncthreads();
    for (int j = threadIdx.x; j < NP; j += 128) {
        for (int i = j; i >= 0; --i) {
            float s = (i == j) ? 1.0f : 0.0f;
            for (int k = i + 1; k <= j; ++k) s -= U[i][k] * Xc[k][j];
            Xc[i][j] = s / U[i][i];
        }
        for (int i = j + 1; i < NP; ++i) Xc[i][j] = 0.0f;
    }
    __syncthreads();
    for (int e = threadIdx.x; e < NPP; e += 128) R[e] = Xc[e / NP][e % NP];
}

// ---------------- WMMA fp32 GEMM kernels ----------------

// batched 80x80x80: C_l = A_l @ B_l (strides in floats; 0 => shared operand)
// operands staged global->LDS via async copies (fully in-bounds by padding)
__global__ __launch_bounds__(256)
void k_bgemm80(const float* __restrict__ A, long long sA,
               const float* __restrict__ B, long long sB,
               float* __restrict__ C, long long sC) {
    __shared__ float As[NP][NP + 1];
    __shared__ float Bs[NP][NP + 1];
    long long l = blockIdx.x;
    const float* Ab = A + l * sA;
    const float* Bb = B + l * sB;
    float*       Cb = C + l * sC;
    for (int e = threadIdx.x; e < NPP; e += 256) {
        int i = e / NP, j = e % NP;
        async_g2l_b32(lds_off(&As[i][j]), Ab + e);
        async_g2l_b32(lds_off(&Bs[i][j]), Bb + e);
    }
    wait_async0();
    __syncthreads();
    int lane = threadIdx.x & 31, wid = threadIdx.x >> 5;
    int l15 = lane & 15, klo = (lane >> 4) * 2, rof = (lane >> 4) * 8;
    for (int t = wid; t < 25; t += 8) {
        int ti = t / 5, tj = t % 5;
        v8f acc = {};
        for (int kk = 0; kk < 20; ++kk) {
            int k0 = kk * 4;
            v2f a, b;
            a.x = As[ti * 16 + l15][k0 + klo];
            a.y = As[ti * 16 + l15][k0 + klo + 1];
            b.x = Bs[k0 + klo][tj * 16 + l15];
            b.y = Bs[k0 + klo + 1][tj * 16 + l15];
            acc = wmma4(a, b, acc);
        }
        for (int r = 0; r < 8; ++r)
            Cb[(ti * 16 + r + rof) * NP + tj * 16 + l15] = acc[r];
    }
}

// big tiled GEMM: C = beta*C + alpha * A(MxK) * op(B)(KxN), row-major
// block tile 64x128, 8 waves of 32x32, BK=16.
// Interior blocks & full-K stages stage tiles with async global->LDS copies;
// ragged edges use guarded VALU loads (zero fill).
__global__ __launch_bounds__(256)
void k_gemm_tiled(const float* __restrict__ A, int lda,
                  const float* __restrict__ B, int ldb,
                  float* __restrict__ C, int ldc,
                  int Mm, int Nn, int Kk, float alpha, float beta, int transB) {
    __shared__ float As[64][17];
    __shared__ float Bs[16][129];
    int bM = blockIdx.x * 64, bN = blockIdx.y * 128;
    int tid = threadIdx.x, lane = tid & 31, wid = tid >> 5;
    int l15 = lane & 15, klo = (lane >> 4) * 2, rof = (lane >> 4) * 8;
    int mb = (wid >> 2) * 32, nb = (wid & 3) * 32;
    bool interior = (bM + 64 <= Mm) && (bN + 128 <= Nn);
    v8f acc[2][2] = {};
    int ks = (Kk + 15) / 16;
    for (int s = 0; s < ks; ++s) {
        int kk0 = s * 16;
        if (interior && (kk0 + 16 <= Kk)) {
#pragma unroll
            for (int q = 0; q < 4; ++q) {
                int e = tid + q * 256;
                int m = e >> 4, k = e & 15;
                async_g2l_b32(lds_off(&As[m][k]),
                              A + (size_t)(bM + m) * lda + (kk0 + k));
            }
#pragma unroll
            for (int q = 0; q < 8; ++q) {
                int e = tid + q * 256;
                int k = e >> 7, n = e & 127;
                const float* gp = transB ? (B + (size_t)(bN + n) * ldb + (kk0 + k))
                                         : (B + (size_t)(kk0 + k) * ldb + (bN + n));
                async_g2l_b32(lds_off(&Bs[k][n]), gp);
            }
            wait_async0();
        } else {
#pragma unroll
            for (int q = 0; q < 4; ++q) {
                int e = tid + q * 256;
                int m = e >> 4, k = e & 15;
                int gm = bM + m, gk = kk0 + k;
                As[m][k] = (gm < Mm && gk < Kk) ? A[(size_t)gm * lda + gk] : 0.0f;
            }
#pragma unroll
            for (int q = 0; q < 8; ++q) {
                int e = tid + q * 256;
                int k = e >> 7, n = e & 127;
                int gk = kk0 + k, gn = bN + n;
                float v = 0.0f;
                if (gk < Kk && gn < Nn)
                    v = transB ? B[(size_t)gn * ldb + gk] : B[(size_t)gk * ldb + gn];
                Bs[k][n] = v;
            }
        }
        __syncthreads();
#pragma unroll
        for (int kc = 0; kc < 4; ++kc) {
            int k0 = kc * 4;
            v2f a0, a1, b0, b1;
            a0.x = As[mb + l15][k0 + klo];      a0.y = As[mb + l15][k0 + klo + 1];
            a1.x = As[mb + 16 + l15][k0 + klo]; a1.y = As[mb + 16 + l15][k0 + klo + 1];
            b0.x = Bs[k0 + klo][nb + l15];      b0.y = Bs[k0 + klo + 1][nb + l15];
            b1.x = Bs[k0 + klo][nb + 16 + l15]; b1.y = Bs[k0 + klo + 1][nb + 16 + l15];
            acc[0][0] = wmma4(a0, b0, acc[0][0]);
            acc[0][1] = wmma4(a0, b1, acc[0][1]);
            acc[1][0] = wmma4(a1, b0, acc[1][0]);
            acc[1][1] = wmma4(a1, b1, acc[1][1]);
        }
        __syncthreads();
    }
#pragma unroll
    for (int mi = 0; mi < 2; ++mi)
#pragma unroll
        for (int ni = 0; ni < 2; ++ni)
#pragma unroll
            for (int r = 0; r < 8; ++r) {
                int gr = bM + mb + mi * 16 + r + rof;
                int gc = bN + nb + ni * 16 + l15;
                if (gr < Mm && gc < Nn) {
                    size_t off = (size_t)gr * ldc + gc;
                    float prev = (beta != 0.0f) ? C[off] : 0.0f;
                    C[off] = beta * prev + alpha * acc[mi][ni][r];
                }
            }
}

// ---------------- batched skmult helpers ----------------

__global__ void k_batched_smat(const float* __restrict__ A, int l0, int lcnt,
                               float* __restrict__ Sb) {
    int idx = blockIdx.x * blockDim.x + threadIdx.x;
    if (idx >= lcnt * NPP) return;
    int l = idx / NPP, r = idx % NPP;
    int i = r / NP, j = r % NP;
    const float* row = A + (size_t)(l0 + l) * NT;
    float val = 0.0f;
    if (i < N70 && j < N70) {
        int a = i > j ? i : j, b = i > j ? j : i;
        float x = row[a * (a + 1) / 2 + b];
        val = (i == j) ? x : SQHF * x;
    }
    Sb[idx] = val;
}

__global__ void k_batched_svec(const float* __restrict__ Wb, int l0, int lcnt,
                               float* __restrict__ SK) {
    int idx = blockIdx.x * blockDim.x + threadIdx.x;
    if (idx >= lcnt * NT) return;
    int l = idx / NT, t = idx % NT;
    int i, j; tril_ij(t, i, j);
    const float* Wm = Wb + (size_t)l * NPP;
    float val = 0.5f * (Wm[i * NP + j] + Wm[j * NP + i]);
    SK[(size_t)t * MD + (l0 + l)] = (i > j ? SQ2F : 1.0f) * val;
}

// out[t] = svecfac * ( sgn*hs(W1) + sgn*hs(W2) + (i==j ? ws[smu_slot] : 0) )
__global__ void k_svec_combo(float* __restrict__ out,
                             const float* __restrict__ W1,
                             const float* __restrict__ W2,
                             const float* __restrict__ ws, int smu_slot,
                             float sgn) {
    int t = blockIdx.x * blockDim.x + threadIdx.x;
    if (t >= NT) return;
    int i, j; tril_ij(t, i, j);
    float val = 0.0f;
    if (W1) val += sgn * 0.5f * (W1[i * NP + j] + W1[j * NP + i]);
    if (W2) val += sgn * 0.5f * (W2[i * NP + j] + W2[j * NP + i]);
    if (smu_slot >= 0 && i == j) val += ws[smu_slot];
    out[t] = (i > j ? SQ2F : 1.0f) * val;
}

// ---------------- blocked Cholesky of Mmat (lower) ----------------

__global__ __launch_bounds__(256)
void k_chol_diag(float* __restrict__ Mp, int p0, int psz) {
    __shared__ float T[64][65];
    for (int e = threadIdx.x; e < psz * psz; e += 256)
        T[e / psz][e % psz] = Mp[(size_t)(p0 + e / psz) * MD + p0 + e % psz];
    __syncthreads();
    for (int k = 0; k < psz; ++k) {
        if (threadIdx.x == 0) T[k][k] = sqrtf(fmaxf(T[k][k], 1e-30f));
        __syncthreads();
        float dk = T[k][k];
        for (int i = k + 1 + threadIdx.x; i < psz; i += 256) T[i][k] /= dk;
        __syncthreads();
        for (int e = threadIdx.x; e < psz * psz; e += 256) {
            int i = e / psz, j = e % psz;
            if (j > k && i >= j) T[i][j] -= T[i][k] * T[j][k];
        }
        __syncthreads();
    }
    for (int e = threadIdx.x; e < psz * psz; e += 256) {
        int i = e / psz, j = e % psz;
        if (i >= j) Mp[(size_t)(p0 + i) * MD + p0 + j] = T[i][j];
    }
}

// panel solve: rows of L21 solve x * L11^T = a (forward substitution)
__global__ __launch_bounds__(128)
void k_trsm(float* __restrict__ Mp, int p0, int psz, int rstart, int rem) {
    __shared__ float L11[64][65];
    __shared__ float Xr[128][65];
    for (int e = threadIdx.x; e < psz * psz; e += 128)
        L11[e / psz][e % psz] = Mp[(size_t)(p0 + e / psz) * MD + p0 + e % psz];
    __syncthreads();
    int grow = rstart + blockIdx.x * 128 + threadIdx.x;
    if (grow < MD) {
        float* row = Mp + (size_t)grow * MD + p0;
        for (int j = 0; j < psz; ++j) Xr[threadIdx.x][j] = row[j];
        for (int j = 0; j < psz; ++j) {
            float s = Xr[threadIdx.x][j];
            for (int k = 0; k < j; ++k) s -= Xr[threadIdx.x][k] * L11[j][k];
            Xr[threadIdx.x][j] = s / L11[j][j];
        }
        for (int j = 0; j < psz; ++j) row[j] = Xr[threadIdx.x][j];
    }
}

__global__ __launch_bounds__(256)
void k_trsolve_fwd(const float* __restrict__ L, const float* __restrict__ b,
                   float* __restrict__ y) {
    __shared__ float r[MD];
    __shared__ float yk;
    for (int i = threadIdx.x; i < MD; i += 256) r[i] = b[i];
    __syncthreads();
    for (int k = 0; k < MD; ++k) {
        if (threadIdx.x == 0) {
            float v = r[k] / L[(size_t)k * MD + k];
            yk = v; y[k] = v;
        }
        __syncthreads();
        float v = yk;
        for (int i = k + 1 + threadIdx.x; i < MD; i += 256)
            r[i] -= L[(size_t)i * MD + k] * v;
        __syncthreads();
    }
}

__global__ __launch_bounds__(256)
void k_trsolve_bwd(const float* __restrict__ L, const float* __restrict__ b,
                   float* __restrict__ y) {
    __shared__ float r[MD];
    __shared__ float yk;
    for (int i = threadIdx.x; i < MD; i += 256) r[i] = b[i];
    __syncthreads();
    for (int k = MD - 1; k >= 0; --k) {
        if (threadIdx.x == 0) {
            float v = r[k] / L[(size_t)k * MD + k];
            yk = v; y[k] = v;
        }
        __syncthreads();
        float v = yk;
        for (int i = threadIdx.x; i < k; i += 256)
            r[i] -= L[(size_t)k * MD + i] * v;
        __syncthreads();
    }
}

// ---------------- GEMV / elementwise / scalar kernels ----------------

// y[i] = add[i] + alpha * dot(A_row_i, x); one wave per row
__global__ void k_gemv_row(const float* __restrict__ A, const float* __restrict__ x,
                           const float* __restrict__ add, float* __restrict__ y,
                           int m, int n, float alpha) {
    int wid = (blockIdx.x * blockDim.x + threadIdx.x) >> 5;
    int lane = threadIdx.x & 31;
    if (wid >= m) return;
    const float* row = A + (size_t)wid * n;
    float s = 0.0f;
    for (int j = lane; j < n; j += 32) s += row[j] * x[j];
    for (int off = 16; off > 0; off >>= 1) s += __shfl_down(s, off, 32);
    if (lane == 0) y[wid] = (add ? add[wid] : 0.0f) + alpha * s;
}

// out[j] = base[j] + alpha * sum_i A[i][j] * v[i]
__global__ void k_gemv_col(const float* __restrict__ A, const float* __restrict__ v,
                           const float* __restrict__ base, float* __restrict__ out,
                           int m, int n, float alpha) {
    int j = blockIdx.x * blockDim.x + threadIdx.x;
    if (j >= n) return;
    float s = 0.0f;
    for (int i = 0; i < m; ++i) s += A[(size_t)i * n + j] * v[i];
    out[j] = (base ? base[j] : 0.0f) + alpha * s;
}

__global__ void k_comb2(float* __restrict__ out, const float* __restrict__ a,
                        float ca, const float* __restrict__ b, float cb, int n) {
    int i = blockIdx.x * blockDim.x + threadIdx.x;
    if (i < n) out[i] = ca * a[i] + cb * b[i];
}

__global__ void k_comb2_slot(float* __restrict__ out, const float* __restrict__ a,
                             const float* __restrict__ b,
                             const float* __restrict__ ws, int slot, int n) {
    int i = blockIdx.x * blockDim.x + threadIdx.x;
    if (i < n) out[i] = a[i] + ws[slot] * b[i];
}

__global__ void k_scale_ptr(float* __restrict__ out, const float* __restrict__ a,
                            const float* __restrict__ sc, int n) {
    int i = blockIdx.x * blockDim.x + threadIdx.x;
    if (i < n) out[i] = a[i] * sc[0];
}

__global__ __launch_bounds__(256)
void k_dot(const float* __restrict__ a, const float* __restrict__ b, int n,
           float* __restrict__ ws, int slot, float scale) {
    __shared__ float red[256];
    float s = 0.0f;
    for (int i = threadIdx.x; i < n; i += 256) s += a[i] * b[i];
    red[threadIdx.x] = s;
    __syncthreads();
    for (int o = 128; o > 0; o >>= 1) {
        if (threadIdx.x < o) red[threadIdx.x] += red[threadIdx.x + o];
        __syncthreads();
    }
    if (threadIdx.x == 0) ws[slot] = scale * red[0];
}

__global__ __launch_bounds__(256)
void k_dot_step(const float* __restrict__ x, const float* __restrict__ dx,
                const float* __restrict__ z, const float* __restrict__ dz,
                float* __restrict__ ws) {
    __shared__ float red[256];
    float al = ws[S_ALPHA], be = ws[S_BETA];
    float s = 0.0f;
    for (int i = threadIdx.x; i < NT; i += 256)
        s += (x[i] + al * dx[i]) * (z[i] + be * dz[i]);
    red[threadIdx.x] = s;
    __syncthreads();
    for (int o = 128; o > 0; o >>= 1) {
        if (threadIdx.x < o) red[threadIdx.x] += red[threadIdx.x + o];
        __syncthreads();
    }
    if (threadIdx.x == 0) ws[S_DOTNEW] = red[0];
}

__global__ void k_mu(float* ws) {
    if (threadIdx.x == 0) ws[S_MU] = ws[S_DOTXZ] / 70.0f;
}

__global__ void k_post_ab(float* ws) {
    if (threadIdx.x) return;
    float mn = fminf(ws[S_ALPHA], ws[S_BETA]);
    ws[S_MINAB] = mn;
    float e = 1.0f;
    if (ws[S_MU] > 1e-6f && mn >= sqrtf(1.0f / 3.0f)) e = fmaxf(1.0f, 3.0f * mn * mn);
    ws[S_E] = e;
    ws[S_G2] = 0.9f + 0.09f * mn;
}

__global__ void k_sigma(float* ws) {
    if (threadIdx.x) return;
    float dn = ws[S_DOTNEW];
    float frac = dn / ws[S_DOTXZ];
    float sig = (dn < 0.0f) ? 0.8f : fminf(1.0f, powf(frac, ws[S_E]));
    ws[S_SIGMA] = sig;
    ws[S_SMU] = sig * ws[S_MU];
}

// cyclic Jacobi, min eigenvalue of 70x70 block of Bin (ld NP), then steplen
__global__ __launch_bounds__(128)
void k_mineig_step(const float* __restrict__ Bin, float* __restrict__ ws,
                   int gslot, float gconst, int outslot) {
    __shared__ float B[N70][N70 + 1];
    __shared__ float cs[2];
    for (int e = threadIdx.x; e < N70 * N70; e += 128)
        B[e / N70][e % N70] = Bin[(e / N70) * NP + (e % N70)];
    __syncthreads();
    for (int sweep = 0; sweep < 8; ++sweep)
        for (int p = 0; p < N70 - 1; ++p)
            for (int q = p + 1; q < N70; ++q) {
                if (threadIdx.x == 0) {
                    float apq = B[p][q];
                    if (fabsf(apq) > 1e-12f) {
                        float th = (B[q][q] - B[p][p]) / (2.0f * apq);
                        float tt = (th >= 0.0f ? 1.0f : -1.0f) /
                                   (fabsf(th) + sqrtf(1.0f + th * th));
                        float c = rsqrtf(1.0f + tt * tt);
                        cs[0] = c; cs[1] = tt * c;
                    } else { cs[0] = 1.0f; cs[1] = 0.0f; }
                }
                __syncthreads();
                float c = cs[0], s = cs[1];
                if (s != 0.0f) {
                    for (int i = threadIdx.x; i < N70; i += 128) {
                        float bip = B[i][p], biq = B[i][q];
                        B[i][p] = c * bip - s * biq;
                        B[i][q] = s * bip + c * biq;
                    }
                    __syncthreads();
                    for (int i = threadIdx.x; i < N70; i += 128) {
                        float bpi = B[p][i], bqi = B[q][i];
                        B[p][i] = c * bpi - s * bqi;
                        B[q][i] = s * bpi + c * bqi;
                    }
                }
                __syncthreads();
            }
    if (threadIdx.x == 0) {
        float lam = B[0][0];
        for (int i = 1; i < N70; ++i) lam = fminf(lam, B[i][i]);
        float sfac = (lam < 0.0f) ? (-1.0f / lam) : 20.0f;
        float g = (gslot >= 0) ? ws[gslot] : gconst;
        ws[outslot] = fminf(0.5f, g * sfac);
    }
}

// ---------------- host orchestration ----------------

extern "C" void kernel_launch(void* const* d_in, const int* in_sizes, int n_in,
                              void* d_out, int out_size, void* d_ws, size_t ws_size,
                              hipStream_t stream) {
    const float* x     = (const float*)d_in[0];
    const float* y     = (const float*)d_in[1];
    const float* z     = (const float*)d_in[2];
    const float* A     = (const float*)d_in[3];
    const float* b     = (const float*)d_in[4];
    const float* c     = (const float*)d_in[5];
    const float* normc = (const float*)d_in[6];
    float* out = (float*)d_out;
    float* w   = (float*)d_ws;
    auto SM = [&](int i) { return w + OFF_SM + (size_t)i * NPP; };
    dim3 B256(256);
    auto g1 = [](int n) { return dim3((unsigned)((n + 255) / 256)); };

    // X, Z (pad diag 1 so Cholesky of padded block is well-defined)
    k_smat80<<<g1(NPP), B256, 0, stream>>>(x, SM(MX), 1.0f);
    k_smat80<<<g1(NPP), B256, 0, stream>>>(z, SM(MZ), 1.0f);
    k_chol_upper80<<<1, 256, 0, stream>>>(SM(MX), SM(MQ));
    k_chol_upper80<<<1, 256, 0, stream>>>(SM(MZ), SM(MP_));
    k_trinv_upper80<<<1, 128, 0, stream>>>(SM(MQ), SM(MQI));
    k_trinv_upper80<<<1, 128, 0, stream>>>(SM(MP_), SM(MPI));
    k_tr80<<<g1(NPP), B256, 0, stream>>>(SM(MQI), SM(MQIT));
    k_tr80<<<g1(NPP), B256, 0, stream>>>(SM(MPI), SM(MPIT));
    // Zinv = Pinv @ Pinv^T
    k_bgemm80<<<1, 256, 0, stream>>>(SM(MPI), 0, SM(MPIT), 0, SM(MZI), 0);

    // rp = b - A x ; cz = c - z ; Rd = cz - A^T y
    k_gemv_row<<<dim3((MD * 32 + 255) / 256), B256, 0, stream>>>(A, x, b, w + V_RP, MD, NT, -1.0f);
    k_comb2<<<g1(NT), B256, 0, stream>>>(w + V_CZ, c, 1.0f, z, -1.0f, NT);
    k_gemv_col<<<g1(NT), B256, 0, stream>>>(A, y, w + V_CZ, w + V_RD, MD, NT, -1.0f);
    k_dot<<<1, 256, 0, stream>>>(x, z, NT, w, S_DOTXZ, 1.0f);
    k_mu<<<1, 32, 0, stream>>>(w);

    // W1 = P @ (X@Z) @ Pinv ; Rc = -svec(halfsym(W1))
    k_bgemm80<<<1, 256, 0, stream>>>(SM(MX), 0, SM(MZ), 0, SM(MXZ), 0);
    k_bgemm80<<<1, 256, 0, stream>>>(SM(MP_), 0, SM(MXZ), 0, SM(MT1), 0);
    k_bgemm80<<<1, 256, 0, stream>>>(SM(MT1), 0, SM(MPI), 0, SM(MW1), 0);
    k_svec_combo<<<g1(NT), B256, 0, stream>>>(w + V_RC, SM(MW1), nullptr, w, -1, -1.0f);

    // SK1 = skmult(X, Zinv, A^T): W_l = Zinv @ smat(A_l) @ X, chunked batches
    for (int ch = 0; ch < MD / CHUNK; ++ch) {
        int l0 = ch * CHUNK;
        k_batched_smat<<<g1(CHUNK * NPP), B256, 0, stream>>>(A, l0, CHUNK, w + OFF_SB);
        k_bgemm80<<<CHUNK, 256, 0, stream>>>(SM(MZI), 0, w + OFF_SB, NPP, w + OFF_UB, NPP);
        k_bgemm80<<<CHUNK, 256, 0, stream>>>(w + OFF_UB, NPP, SM(MX), 0, w + OFF_WB, NPP);
        k_batched_svec<<<g1(CHUNK * NT), B256, 0, stream>>>(w + OFF_WB, l0, CHUNK, w + OFF_SK);
    }

    // Mmat = A @ SK1  (compute-bound: fp32 WMMA tiled GEMM)
    {
        dim3 gmm((MD + 63) / 64, (MD + 127) / 128);
        k_gemm_tiled<<<gmm, B256, 0, stream>>>(A, NT, w + OFF_SK, MD, w + OFF_MM, MD,
                                               MD, MD, NT, 1.0f, 0.0f, 0);
    }
    // blocked Cholesky of Mmat (lower), trailing updates via WMMA GEMM (-L21 L21^T)
    for (int p0 = 0; p0 < MD; p0 += 64) {
        int psz = (MD - p0) < 64 ? (MD - p0) : 64;
        k_chol_diag<<<1, 256, 0, stream>>>(w + OFF_MM, p0, psz);
        int rstart = p0 + psz, rem = MD - rstart;
        if (rem > 0) {
            k_trsm<<<dim3((rem + 127) / 128), dim3(128), 0, stream>>>(w + OFF_MM, p0, psz, rstart, rem);
            dim3 gu((rem + 63) / 64, (rem + 127) / 128);
            k_gemm_tiled<<<gu, B256, 0, stream>>>(
                w + OFF_MM + (size_t)rstart * MD + p0, MD,
                w + OFF_MM + (size_t)rstart * MD + p0, MD,
                w + OFF_MM + (size_t)rstart * MD + rstart, MD,
                rem, rem, psz, -1.0f, 1.0f, 1);
        }
    }

    // u1 = skmult(X,Zinv,Rd)
    k_smat80<<<g1(NPP), B256, 0, stream>>>(w + V_RD, SM(MS), 0.0f);
    k_bgemm80<<<1, 256, 0, stream>>>(SM(MZI), 0, SM(MS), 0, SM(MT1), 0);
    k_bgemm80<<<1, 256, 0, stream>>>(SM(MT1), 0, SM(MX), 0, SM(MT2), 0);
    k_svec_combo<<<g1(NT), B256, 0, stream>>>(w + V_U1, SM(MT2), nullptr, w, -1, 1.0f);
    // u2 = skmult(Pinv,Pinv,Rc)
    k_smat80<<<g1(NPP), B256, 0, stream>>>(w + V_RC, SM(MS), 0.0f);
    k_bgemm80<<<1, 256, 0, stream>>>(SM(MPI), 0, SM(MS), 0, SM(MT1), 0);
    k_bgemm80<<<1, 256, 0, stream>>>(SM(MT1), 0, SM(MPIT), 0, SM(MT2), 0);
    k_svec_combo<<<g1(NT), B256, 0, stream>>>(w + V_U2, SM(MT2), nullptr, w, -1, 1.0f);
    // h = rp + A @ (u1 - u2); dy = M^{-1} h
    k_comb2<<<g1(NT), B256, 0, stream>>>(w + V_TV, w + V_U1, 1.0f, w + V_U2, -1.0f, NT);
    k_gemv_row<<<dim3((MD * 32 + 255) / 256), B256, 0, stream>>>(A, w + V_TV, w + V_RP, w + V_H, MD, NT, 1.0f);
    k_trsolve_fwd<<<1, 256, 0, stream>>>(w + OFF_MM, w + V_H, w + V_YT);
    k_trsolve_bwd<<<1, 256, 0, stream>>>(w + OFF_MM, w + V_YT, w + V_DY);
    // dz = Rd - A^T dy ; dx = -x - skmult(X,Zinv,dz)
    k_gemv_col<<<g1(NT), B256, 0, stream>>>(A, w + V_DY, w + V_RD, w + V_DZ, MD, NT, -1.0f);
    k_smat80<<<g1(NPP), B256, 0, stream>>>(w + V_DZ, SM(MS), 0.0f);
    k_bgemm80<<<1, 256, 0, stream>>>(SM(MZI), 0, SM(MS), 0, SM(MT1), 0);
    k_bgemm80<<<1, 256, 0, stream>>>(SM(MT1), 0, SM(MX), 0, SM(MT2), 0);
    k_svec_combo<<<g1(NT), B256, 0, stream>>>(w + V_U4, SM(MT2), nullptr, w, -1, 1.0f);
    k_comb2<<<g1(NT), B256, 0, stream>>>(w + V_DX, x, -1.0f, w + V_U4, -1.0f, NT);

    // alpha = steplen(dx, Qinv, 0.9): min-eig of Qinv^T smat(dx) Qinv
    k_smat80<<<g1(NPP), B256, 0, stream>>>(w + V_DX, SM(MS), 0.0f);
    k_bgemm80<<<1, 256, 0, stream>>>(SM(MQIT), 0, SM(MS), 0, SM(MT1), 0);
    k_bgemm80<<<1, 256, 0, stream>>>(SM(MT1), 0, SM(MQI), 0, SM(MB), 0);
    k_mineig_step<<<1, 128, 0, stream>>>(SM(MB), w, -1, 0.9f, S_ALPHA);
    // beta = steplen(dz, Pinv, 0.9)
    k_smat80<<<g1(NPP), B256, 0, stream>>>(w + V_DZ, SM(MS), 0.0f);
    k_bgemm80<<<1, 256, 0, stream>>>(SM(MPIT), 0, SM(MS), 0, SM(MT1), 0);
    k_bgemm80<<<1, 256, 0, stream>>>(SM(MT1), 0, SM(MPI), 0, SM(MB), 0);
    k_mineig_step<<<1, 128, 0, stream>>>(SM(MB), w, -1, 0.9f, S_BETA);

    k_post_ab<<<1, 32, 0, stream>>>(w);
    k_dot_step<<<1, 256, 0, stream>>>(x, w + V_DX, z, w + V_DZ, w);
    k_sigma<<<1, 32, 0, stream>>>(w);

    // Rc2 = svec(sigma*mu*I - Hp(XZ) - Hp(smat(dx)@smat(dz)))
    k_smat80<<<g1(NPP), B256, 0, stream>>>(w + V_DX, SM(MS), 0.0f);
    k_smat80<<<g1(NPP), B256, 0, stream>>>(w + V_DZ, SM(MT2), 0.0f);
    k_bgemm80<<<1, 256, 0, stream>>>(SM(MS), 0, SM(MT2), 0, SM(MD2), 0);
    k_bgemm80<<<1, 256, 0, stream>>>(SM(MP_), 0, SM(MD2), 0, SM(MT1), 0);
    k_bgemm80<<<1, 256, 0, stream>>>(SM(MT1), 0, SM(MPI), 0, SM(MW2), 0);
    k_svec_combo<<<g1(NT), B256, 0, stream>>>(w + V_RC, SM(MW1), SM(MW2), w, S_SMU, -1.0f);

    // u3 = skmult(Pinv,Pinv,Rc2)
    k_smat80<<<g1(NPP), B256, 0, stream>>>(w + V_RC, SM(MS), 0.0f);
    k_bgemm80<<<1, 256, 0, stream>>>(SM(MPI), 0, SM(MS), 0, SM(MT1), 0);
    k_bgemm80<<<1, 256, 0, stream>>>(SM(MT1), 0, SM(MPIT), 0, SM(MT2), 0);
    k_svec_combo<<<g1(NT), B256, 0, stream>>>(w + V_U3, SM(MT2), nullptr, w, -1, 1.0f);
    // h2 = rp + A@(u1-u3); dy2 = M^{-1} h2; dz2 = Rd - A^T dy2
    k_comb2<<<g1(NT), B256, 0, stream>>>(w + V_TV, w + V_U1, 1.0f, w + V_U3, -1.0f, NT);
    k_gemv_row<<<dim3((MD * 32 + 255) / 256), B256, 0, stream>>>(A, w + V_TV, w + V_RP, w + V_H2, MD, NT, 1.0f);
    k_trsolve_fwd<<<1, 256, 0, stream>>>(w + OFF_MM, w + V_H2, w + V_YT);
    k_trsolve_bwd<<<1, 256, 0, stream>>>(w + OFF_MM, w + V_YT, w + V_DY2);
    k_gemv_col<<<g1(NT), B256, 0, stream>>>(A, w + V_DY2, w + V_RD, w + V_DZ2, MD, NT, -1.0f);
    // dx2 = u3 - skmult(X,Zinv,dz2)
    k_smat80<<<g1(NPP), B256, 0, stream>>>(w + V_DZ2, SM(MS), 0.0f);
    k_bgemm80<<<1, 256, 0, stream>>>(SM(MZI), 0, SM(MS), 0, SM(MT1), 0);
    k_bgemm80<<<1, 256, 0, stream>>>(SM(MT1), 0, SM(MX), 0, SM(MT2), 0);
    k_svec_combo<<<g1(NT), B256, 0, stream>>>(w + V_U5, SM(MT2), nullptr, w, -1, 1.0f);
    k_comb2<<<g1(NT), B256, 0, stream>>>(w + V_DX2, w + V_U3, 1.0f, w + V_U5, -1.0f, NT);

    // alpha2 / beta2 with g2 from slot
    k_smat80<<<g1(NPP), B256, 0, stream>>>(w + V_DX2, SM(MS), 0.0f);
    k_bgemm80<<<1, 256, 0, stream>>>(SM(MQIT), 0, SM(MS), 0, SM(MT1), 0);
    k_bgemm80<<<1, 256, 0, stream>>>(SM(MT1), 0, SM(MQI), 0, SM(MB), 0);
    k_mineig_step<<<1, 128, 0, stream>>>(SM(MB), w, S_G2, 0.0f, S_ALPHA2);
    k_smat80<<<g1(NPP), B256, 0, stream>>>(w + V_DZ2, SM(MS), 0.0f);
    k_bgemm80<<<1, 256, 0, stream>>>(SM(MPIT), 0, SM(MS), 0, SM(MT1), 0);
    k_bgemm80<<<1, 256, 0, stream>>>(SM(MT1), 0, SM(MPI), 0, SM(MB), 0);
    k_mineig_step<<<1, 128, 0, stream>>>(SM(MB), w, S_G2, 0.0f, S_BETA2);

    // outputs: x_new | y_new | z_new | y1
    float* o_x  = out;
    float* o_y  = out + NT;
    float* o_z  = o_y + MD;
    float* o_y1 = o_z + NT;
    k_comb2_slot<<<g1(NT), B256, 0, stream>>>(o_x, x, w + V_DX2, w, S_ALPHA2, NT);
    k_comb2_slot<<<g1(MD), B256, 0, stream>>>(o_y, y, w + V_DY2, w, S_BETA2, MD);
    k_comb2_slot<<<g1(NT), B256, 0, stream>>>(o_z, z, w + V_DZ2, w, S_BETA2, NT);
    k_scale_ptr<<<g1(MD), B256, 0, stream>>>(o_y1, o_y, normc, MD);
    (void)in_sizes; (void)n_in; (void)out_size; (void)ws_size;
}